// ContextEncodingModule_1666447310975
// MI455X (gfx1250) — compile-verified
//
#include <hip/hip_runtime.h>

typedef float v2f __attribute__((ext_vector_type(2)));
typedef float v8f __attribute__((ext_vector_type(8)));

#define FDIM 512
#define KCW  32
#define NPIX 4096          // 64*64
#define BATCH 16
#define FP   516           // padded LDS row stride (floats): bank-conflict-free
#define ROWS_PER_BLOCK 256
#define NTILES (ROWS_PER_BLOCK / 16)

// ---------------------------------------------------------------------------
// CDNA5 async global->LDS helpers (ASYNCcnt-tracked; inline asm is portable
// across ROCm/clang toolchains whose builtin arity differs)
// ---------------------------------------------------------------------------
__device__ __forceinline__ void async_load_b128(unsigned lds_byte_off,
                                                unsigned long long gaddr) {
    asm volatile("global_load_async_to_lds_b128 %0, %1, off"
                 :: "v"(lds_byte_off), "v"(gaddr) : "memory");
}
__device__ __forceinline__ void wait_asynccnt_le8() {
    asm volatile("s_wait_asynccnt 0x8" ::: "memory");
}
__device__ __forceinline__ void wait_asynccnt_0() {
    asm volatile("s_wait_asynccnt 0x0" ::: "memory");
}

// ---------------------------------------------------------------------------
// Kernel 0: zero the accumulator workspace (enc partials + s_sum)
// ---------------------------------------------------------------------------
__global__ void zero_kernel(float* __restrict__ p, long n) {
    for (long i = (long)blockIdx.x * blockDim.x + threadIdx.x; i < n;
         i += (long)gridDim.x * blockDim.x)
        p[i] = 0.0f;
}

// ---------------------------------------------------------------------------
// Kernel 1: encoder main pass.
//  - feats tiles staged by GLOBAL_LOAD_ASYNC_TO_LDS_B128, double-buffered
//  - GEMM1 (xc = feats . cw^T) via v_wmma_f32_16x16x4_f32
//  - softmax over K=32 with smoothing
//  - GEMM2 (enc_part = s^T . feats) via v_wmma_f32_16x16x4_f32, atomicAdd out
// ---------------------------------------------------------------------------
__global__ __launch_bounds__(256) void encoder_kernel(
    const float* __restrict__ inputs,      // (B, N, F)
    const float* __restrict__ codewords,   // (K, F)
    const float* __restrict__ smoothing,   // (K,)
    float* __restrict__ ws_enc,            // (B, K, F) accumulated s^T x
    float* __restrict__ ws_ssum)           // (B, K)    accumulated sum_n s
{
    extern __shared__ float smem[];
    float* s_cw    = smem;                 // KCW * FP
    float* s_feats = s_cw + KCW * FP;      // 2 * 16 * FP   (double buffer)
    float* s_s     = s_feats + 2 * 16 * FP;// 16 * 36
    float* s_xc    = s_s + 16 * 36;        // 16 * 36
    float* s_ssum  = s_xc + 16 * 36;       // 32
    float* s_c2    = s_ssum + 32;          // 32

    const int tid  = threadIdx.x;
    const int lane = tid & 31;
    const int wave = tid >> 5;
    const int half = lane >> 4;            // 0: lanes 0-15, 1: lanes 16-31
    const int l    = lane & 15;

    const int b     = blockIdx.x / (NPIX / ROWS_PER_BLOCK);
    const int chunk = blockIdx.x % (NPIX / ROWS_PER_BLOCK);
    const int n0    = chunk * ROWS_PER_BLOCK;

    const float4* in4 = reinterpret_cast<const float4*>(inputs);

    // issue the 8 async b128 loads for one 16x512 feats tile into buf
    auto issue_tile = [&](int t, float* buf) {
        const long rowg = (long)b * NPIX + n0 + t * 16;
        for (int j = 0; j < 8; ++j) {
            int idx4 = tid + j * 256;
            int r = idx4 >> 7;             // / (FDIM/4)
            int c = idx4 & 127;            // % (FDIM/4)
            unsigned lds_off =
                (unsigned)(uintptr_t)(&buf[r * FP + c * 4]);
            unsigned long long ga =
                (unsigned long long)(uintptr_t)(&in4[(rowg + r) * (FDIM / 4) + c]);
            async_load_b128(lds_off, ga);
        }
    };

    // ---- stage codewords in LDS (32x512 -> 4096 float4, 16 per thread) ----
    {
        const float4* cw4 = reinterpret_cast<const float4*>(codewords);
        for (int j = 0; j < (KCW * FDIM / 4) / 256; ++j) {
            int idx4 = tid + j * 256;
            int r = idx4 / (FDIM / 4);
            int c = idx4 % (FDIM / 4);
            float4 v = cw4[r * (FDIM / 4) + c];
            *reinterpret_cast<float4*>(&s_cw[r * FP + c * 4]) = v;
        }
    }
    issue_tile(0, s_feats);                // prefetch tile 0 into buffer 0
    if (tid < KCW) s_ssum[tid] = 0.0f;
    __syncthreads();
    if (tid < KCW) {                       // ||c_k||^2
        float a = 0.0f;
        for (int f = 0; f < FDIM; ++f) { float v = s_cw[tid * FP + f]; a += v * v; }
        s_c2[tid] = a;
    }

    // persistent GEMM2 accumulators: 2 k-tiles x 4 f-tiles of 16x16 f32
    v8f acc[2][4];
    for (int kt = 0; kt < 2; ++kt)
        for (int ft = 0; ft < 4; ++ft)
            for (int e = 0; e < 8; ++e) acc[kt][ft][e] = 0.0f;

    for (int t = 0; t < NTILES; ++t) {
        if (t) __syncthreads();            // prev GEMM2 done reading buf/s_s

        float* fb_cur = s_feats + (t & 1) * (16 * FP);
        if (t + 1 < NTILES) {              // stream next tile behind compute
            issue_tile(t + 1, s_feats + ((t + 1) & 1) * (16 * FP));
            wait_asynccnt_le8();           // tile t's 8 loads retired
        } else {
            wait_asynccnt_0();
        }
        __syncthreads();                   // tile t visible to all waves

        // ---- GEMM1: xc[r][k] = sum_f feats[r][f] * cw[k][f]  (waves 0,1) ----
        if (wave < 2) {
            const int k0 = wave * 16;
            v8f c1;
            for (int e = 0; e < 8; ++e) c1[e] = 0.0f;
            for (int f0 = 0; f0 < FDIM; f0 += 4) {
                v2f a, bm;
                // A 16x4 layout: v0 = K{0|2}, v1 = K{1|3} split at lane 16
                a.x  = fb_cur[l * FP + f0 + 2 * half];
                a.y  = fb_cur[l * FP + f0 + 2 * half + 1];
                // B 4x16 layout: B[f][k] = cw[k][f]
                bm.x = s_cw[(k0 + l) * FP + f0 + 2 * half];
                bm.y = s_cw[(k0 + l) * FP + f0 + 2 * half + 1];
                c1 = __builtin_amdgcn_wmma_f32_16x16x4_f32(
                        false, a, false, bm, (short)0, c1, false, false);
            }
            // D layout: vgpr j = row (j + 8*half), col l
            for (int j = 0; j < 8; ++j)
                s_xc[(j + 8 * half) * 36 + k0 + l] = c1[j];
        }
        __syncthreads();

        // ---- softmax over K=32 per row (threads 0..15) ----
        if (tid < 16) {
            const int r = tid;
            float x2 = 0.0f;
            for (int f = 0; f < FDIM; ++f) { float v = fb_cur[r * FP + f]; x2 += v * v; }
            float q[KCW];
            float mx = -3.4e38f;
            for (int k = 0; k < KCW; ++k) {
                float sq = x2 - 2.0f * s_xc[r * 36 + k] + s_c2[k];
                q[k] = sq * smoothing[k];
                mx = fmaxf(mx, q[k]);
            }
            float sum = 0.0f;
            for (int k = 0; k < KCW; ++k) { q[k] = __expf(q[k] - mx); sum += q[k]; }
            float inv = 1.0f / sum;
            for (int k = 0; k < KCW; ++k) s_s[r * 36 + k] = q[k] * inv;
        }
        __syncthreads();

        if (tid < KCW) {                   // block-local s_sum accumulation
            float a = 0.0f;
            for (int r = 0; r < 16; ++r) a += s_s[r * 36 + tid];
            s_ssum[tid] += a;
        }

        // ---- GEMM2: enc[k][f] += sum_r s[r][k] * feats[r][f]  (all waves) ----
        const int fb = wave * 64;
        for (int r0 = 0; r0 < 16; r0 += 4) {
            v2f bm[4];
            for (int ft = 0; ft < 4; ++ft) {
                bm[ft].x = fb_cur[(r0 + 2 * half) * FP + fb + ft * 16 + l];
                bm[ft].y = fb_cur[(r0 + 2 * half + 1) * FP + fb + ft * 16 + l];
            }
            for (int kt = 0; kt < 2; ++kt) {
                v2f a;  // A[M=k][Kdim=r] = s[r][k]
                a.x = s_s[(r0 + 2 * half) * 36 + kt * 16 + l];
                a.y = s_s[(r0 + 2 * half + 1) * 36 + kt * 16 + l];
                for (int ft = 0; ft < 4; ++ft)
                    acc[kt][ft] = __builtin_amdgcn_wmma_f32_16x16x4_f32(
                        false, a, false, bm[ft], (short)0, acc[kt][ft], false, false);
            }
        }
    }

    // ---- flush enc partials + s_sum ----
    for (int kt = 0; kt < 2; ++kt)
        for (int ft = 0; ft < 4; ++ft)
            for (int j = 0; j < 8; ++j) {
                int kc = kt * 16 + j + 8 * half;
                int f  = wave * 64 + ft * 16 + l;
                atomicAdd(&ws_enc[((long)b * KCW + kc) * FDIM + f], acc[kt][ft][j]);
            }
    __syncthreads();
    if (tid < KCW) atomicAdd(&ws_ssum[b * KCW + tid], s_ssum[tid]);
}

// ---------------------------------------------------------------------------
// Kernel 2: enc = s^Tx - s_sum*cw ; BN(inference) + ReLU ; sum over K -> (B,F)
// ---------------------------------------------------------------------------
__global__ void bn_reduce_kernel(
    const float* __restrict__ ws_enc, const float* __restrict__ ws_ssum,
    const float* __restrict__ codewords,
    const float* __restrict__ gamma, const float* __restrict__ beta,
    const float* __restrict__ mean, const float* __restrict__ var,
    float* __restrict__ ws_encsum)
{
    const int b = blockIdx.x;
    const int f = threadIdx.x;
    const float inv = rsqrtf(var[f] + 1e-3f) * gamma[f];
    const float mu = mean[f], bt = beta[f];
    float acc = 0.0f;
    for (int k = 0; k < KCW; ++k) {
        float e = ws_enc[((long)b * KCW + k) * FDIM + f]
                - ws_ssum[b * KCW + k] * codewords[k * FDIM + f];
        float ebn = (e - mu) * inv + bt;
        acc += fmaxf(ebn, 0.0f);
    }
    ws_encsum[b * FDIM + f] = acc;
}

// ---------------------------------------------------------------------------
// Kernel 3: attn = sigmoid(enc @ W + b) ; se_loss = enc @ w_se + b_se
// ---------------------------------------------------------------------------
__global__ void fc_kernel(
    const float* __restrict__ ws_encsum,
    const float* __restrict__ fc_enc_w, const float* __restrict__ fc_enc_b,
    const float* __restrict__ fc_se_w, const float* __restrict__ fc_se_b,
    float* __restrict__ ws_attn, float* __restrict__ se_out)
{
    __shared__ float enc_s[FDIM];
    __shared__ float red[FDIM];
    const int b = blockIdx.x;
    const int fo = threadIdx.x;
    enc_s[fo] = ws_encsum[b * FDIM + fo];
    __syncthreads();
    float d = fc_enc_b[fo];
    for (int f = 0; f < FDIM; ++f) d += enc_s[f] * fc_enc_w[f * FDIM + fo];
    ws_attn[b * FDIM + fo] = 1.0f / (1.0f + __expf(-d));

    red[fo] = enc_s[fo] * fc_se_w[fo];
    __syncthreads();
    for (int s = FDIM / 2; s > 0; s >>= 1) {
        if (fo < s) red[fo] += red[fo + s];
        __syncthreads();
    }
    if (fo == 0) se_out[b] = red[0] + fc_se_b[0];
}

// ---------------------------------------------------------------------------
// Kernel 4: featuremaps_out = attn[b,f] * inputs[b,h,w,f]  (float4 streaming)
// ---------------------------------------------------------------------------
__global__ void scale_kernel(
    const float* __restrict__ inputs, const float* __restrict__ ws_attn,
    float* __restrict__ out)
{
    const float4* in4 = reinterpret_cast<const float4*>(inputs);
    const float4* at4 = reinterpret_cast<const float4*>(ws_attn);
    float4* out4 = reinterpret_cast<float4*>(out);
    const long total4 = (long)BATCH * NPIX * FDIM / 4;
    const long stride = (long)gridDim.x * blockDim.x;
    for (long i = (long)blockIdx.x * blockDim.x + threadIdx.x; i < total4; i += stride) {
        long flat = i * 4;
        int f4 = (int)((flat % FDIM) / 4);
        int bb = (int)(flat / ((long)NPIX * FDIM));
        float4 a = at4[bb * (FDIM / 4) + f4];
        float4 v = in4[i];
        float4 r;
        r.x = a.x * v.x; r.y = a.y * v.y; r.z = a.z * v.z; r.w = a.w * v.w;
        out4[i] = r;
    }
}

// ---------------------------------------------------------------------------
extern "C" void kernel_launch(void* const* d_in, const int* in_sizes, int n_in,
                              void* d_out, int out_size, void* d_ws, size_t ws_size,
                              hipStream_t stream) {
    const float* inputs    = (const float*)d_in[0];
    const float* codewords = (const float*)d_in[1];
    const float* smoothing = (const float*)d_in[2];
    const float* bn_gamma  = (const float*)d_in[3];
    const float* bn_beta   = (const float*)d_in[4];
    const float* bn_mean   = (const float*)d_in[5];
    const float* bn_var    = (const float*)d_in[6];
    const float* fc_enc_w  = (const float*)d_in[7];
    const float* fc_enc_b  = (const float*)d_in[8];
    const float* fc_se_w   = (const float*)d_in[9];
    const float* fc_se_b   = (const float*)d_in[10];

    float* out = (float*)d_out;
    float* ws  = (float*)d_ws;
    float* ws_enc    = ws;                                  // B*K*F = 262144
    float* ws_ssum   = ws_enc + (long)BATCH * KCW * FDIM;   // B*K = 512
    float* ws_encsum = ws_ssum + BATCH * KCW;               // B*F = 8192
    float* ws_attn   = ws_encsum + BATCH * FDIM;            // B*F = 8192

    // 0) zero accumulators
    const long nz = (long)BATCH * KCW * FDIM + BATCH * KCW;
    zero_kernel<<<(int)((nz + 255) / 256), 256, 0, stream>>>(ws, nz);

    // 1) encoder (async-LDS double-buffered staging + WMMA f32 GEMMs)
    const size_t smem_bytes =
        (size_t)(KCW * FP + 2 * 16 * FP + 16 * 36 + 16 * 36 + 32 + 32) * sizeof(float);
    encoder_kernel<<<BATCH * (NPIX / ROWS_PER_BLOCK), 256, smem_bytes, stream>>>(
        inputs, codewords, smoothing, ws_enc, ws_ssum);

    // 2) BN + ReLU + sum over codewords
    bn_reduce_kernel<<<BATCH, FDIM, 0, stream>>>(
        ws_enc, ws_ssum, codewords, bn_gamma, bn_beta, bn_mean, bn_var, ws_encsum);

    // 3) FC layers: attention + se_loss (se_loss lives after featuremaps in d_out)
    fc_kernel<<<BATCH, FDIM, 0, stream>>>(
        ws_encsum, fc_enc_w, fc_enc_b, fc_se_w, fc_se_b,
        ws_attn, out + (long)BATCH * NPIX * FDIM);

    // 4) streaming rescale of the input tensor
    scale_kernel<<<4096, 256, 0, stream>>>(inputs, ws_attn, out);
}